// PnP3D_13116830122723
// MI455X (gfx1250) — compile-verified
//
#include <hip/hip_runtime.h>
#include <hip/hip_bf16.h>

typedef float v2f __attribute__((ext_vector_type(2)));
typedef float v8f __attribute__((ext_vector_type(8)));

#define CDIM 512
#define NPTS 8192
#define NB   4
#define KNN  16

// ---------------------------------------------------------------------------
// Prep 1: pack xyz (B,3,N) -> float4 (x, y, z, x^2+y^2+z^2), per batch-point.
// ---------------------------------------------------------------------------
__global__ __launch_bounds__(256) void prep_pts_kernel(const float* __restrict__ xyz,
                                                       float4* __restrict__ pts) {
    int t = blockIdx.x * 256 + threadIdx.x;            // 0 .. B*N-1
    if (t >= NB * NPTS) return;
    int b = t >> 13;                                   // t / 8192
    int n = t & (NPTS - 1);
    const float* base = xyz + (size_t)b * 3 * NPTS;
    float x = base[n];
    float y = base[NPTS + n];
    float z = base[2 * NPTS + n];
    pts[t] = make_float4(x, y, z, x * x + y * y + z * z);
}

// ---------------------------------------------------------------------------
// Prep 2: fold BN into per-channel scale/shift: s = gamma*rsqrt(var+eps),
// t = beta - mean*s.  Layout in ws: sl[512] tl[512] sg[512] tg[512].
// ---------------------------------------------------------------------------
__global__ __launch_bounds__(256) void prep_scales_kernel(
    const float* __restrict__ gl, const float* __restrict__ bl,
    const float* __restrict__ ml, const float* __restrict__ vl,
    const float* __restrict__ gg, const float* __restrict__ bg,
    const float* __restrict__ mg, const float* __restrict__ vg,
    float* __restrict__ sc) {
    int c = blockIdx.x * 256 + threadIdx.x;
    if (c >= CDIM) return;
    float s = gl[c] * rsqrtf(vl[c] + 1e-5f);
    sc[c]            = s;
    sc[CDIM + c]     = bl[c] - ml[c] * s;
    float s2 = gg[c] * rsqrtf(vg[c] + 1e-5f);
    sc[2 * CDIM + c] = s2;
    sc[3 * CDIM + c] = bg[c] - mg[c] * s2;
}

// ---------------------------------------------------------------------------
// Fused dual conv1x1 + BN + ReLU + add, via V_WMMA_F32_16X16X4_F32.
// One wave -> one 16(o) x 16(n) tile; both GEMMs share the B operand.
// Block = 8 waves = 8 o-tiles sharing one n-tile (feature loads hit L1).
// Grid: B * (512/16/8) * (8192/16) = 4*4*512 = 8192 blocks.
// A layout (16x4 f32): lane m=l&15; VGPR0/1 = K{0,1} (lanes 0-15), K{2,3} (16-31).
// ---------------------------------------------------------------------------
__global__ __launch_bounds__(256) void conv_fused_kernel(
    const float* __restrict__ feat, const float* __restrict__ wl,
    const float* __restrict__ wg, const float* __restrict__ sc,
    float* __restrict__ out) {
    int lane = threadIdx.x & 31;
    int wv   = threadIdx.x >> 5;
    int nt = blockIdx.x & 511;
    int og = (blockIdx.x >> 9) & 3;
    int b  = blockIdx.x >> 11;
    int ot = og * 8 + wv;
    int mn = lane & 15;
    int h  = lane >> 4;

    const float* arowl = wl + (size_t)(ot * 16 + mn) * CDIM;
    const float* arowg = wg + (size_t)(ot * 16 + mn) * CDIM;
    const float* bcol  = feat + (size_t)b * CDIM * NPTS + nt * 16 + mn;

    v8f accl = {};
    v8f accg = {};
#pragma unroll 4
    for (int kk = 0; kk < CDIM; kk += 4) {
        int k0 = kk + 2 * h;
        v2f al = *(const v2f*)(arowl + k0);
        v2f ag = *(const v2f*)(arowg + k0);
        v2f bv;
        bv.x = bcol[(size_t)k0 * NPTS];
        bv.y = bcol[(size_t)(k0 + 1) * NPTS];
        accl = __builtin_amdgcn_wmma_f32_16x16x4_f32(false, al, false, bv,
                                                     (short)0, accl, false, false);
        accg = __builtin_amdgcn_wmma_f32_16x16x4_f32(false, ag, false, bv,
                                                     (short)0, accg, false, false);
    }

    float* orow = out + (size_t)b * CDIM * NPTS + nt * 16 + mn;
#pragma unroll
    for (int v = 0; v < 8; ++v) {
        int o = ot * 16 + v + 8 * h;
        float yl = fmaxf(accl[v] * sc[o] + sc[CDIM + o], 0.f);
        float yg = fmaxf(accg[v] * sc[2 * CDIM + o] + sc[3 * CDIM + o], 0.f);
        orow[(size_t)o * NPTS] = yl + yg;
    }
}

// ---------------------------------------------------------------------------
// WMMA KNN: one wave handles 16 queries (B columns, loop-invariant operand),
// streams 512 candidate tiles of 16 (A rows).  dist tile = A(16x4) x B(4x16)
// with A=(x,y,z,|c|^2), B=(2qx,2qy,2qz,-1) -> 2 q.c - |c|^2; minus |q|^2/lane.
// Lane (l) and (l+16) see candidate rows v+0..7 / v+8..15 of each tile; each
// keeps a sorted top-16, merged stably at the end via per-wave LDS.
// Grid: B*N/16 waves = 2048 -> 256 blocks of 8 waves.
// ---------------------------------------------------------------------------
__global__ __launch_bounds__(256) void knn_wmma_kernel(const float4* __restrict__ pts,
                                                       int* __restrict__ idxout) {
    __shared__ float sd[256 * KNN];
    __shared__ int   si[256 * KNN];

    int lane = threadIdx.x & 31;
    int wv   = threadIdx.x >> 5;
    int gw = blockIdx.x * 8 + wv;
    int b  = gw >> 9;          // 512 query tiles per batch
    int qt = gw & 511;
    int mn = lane & 15;
    int h  = lane >> 4;

    const float4* P = pts + (size_t)b * NPTS;
    float4 q = P[qt * 16 + mn];
    float xxq = q.w;
    v2f bv;
    bv.x = h ? (2.f * q.z) : (2.f * q.x);
    bv.y = h ? (-1.f)      : (2.f * q.y);

    float bd[KNN];
    int   bi[KNN];
#pragma unroll
    for (int p = 0; p < KNN; ++p) { bd[p] = -3.4e38f; bi[p] = 0; }

    for (int t = 0; t < NPTS / 16; ++t) {
        // A operand: lane loads half of float4 of candidate (t*16+mn): h=0 ->
        // (x,y) = K0,K1 ; h=1 -> (z,|c|^2) = K2,K3.  Coalesced 8B loads.
        v2f a = *((const v2f*)&P[t * 16 + mn] + h);
        v8f acc = {};
        acc = __builtin_amdgcn_wmma_f32_16x16x4_f32(false, a, false, bv,
                                                    (short)0, acc, false, false);
        int jbase = t * 16 + 8 * h;
#pragma unroll
        for (int v = 0; v < 8; ++v) {
            float d = acc[v] - xxq;                 // -||c - q||^2
            if (d > bd[KNN - 1]) {
                float val = d; int vi = jbase + v;
#pragma unroll
                for (int p = 0; p < KNN; ++p) {
                    if (val > bd[p]) {
                        float tv = bd[p]; int ti = bi[p];
                        bd[p] = val; bi[p] = vi;
                        val = tv; vi = ti;
                    }
                }
            }
        }
    }

    // Stage both half-lists in LDS, then lane pair (l, l+16) merges stably.
    int tbase = threadIdx.x * KNN;
#pragma unroll
    for (int p = 0; p < KNN; ++p) { sd[tbase + p] = bd[p]; si[tbase + p] = bi[p]; }
    __syncthreads();

    if (h == 0) {
        int pa = tbase;
        int pb = tbase + 16 * KNN;                  // partner = thread + 16
        int nq = qt * 16 + mn;
        int* orow = idxout + (size_t)(b * NPTS + nq) * KNN;
        int i = 0, j = 0;
        for (int p = 0; p < KNN; ++p) {
            float da = sd[pa + i], db = sd[pb + j];
            int   ia = si[pa + i], ib = si[pb + j];
            bool takeA = (da > db) || ((da == db) && (ia < ib));
            orow[p] = takeA ? ia : ib;
            i += takeA ? 1 : 0;
            j += takeA ? 0 : 1;
        }
    }
}

// ---------------------------------------------------------------------------
extern "C" void kernel_launch(void* const* d_in, const int* in_sizes, int n_in,
                              void* d_out, int out_size, void* d_ws, size_t ws_size,
                              hipStream_t stream) {
    const float* xyz  = (const float*)d_in[0];
    const float* feat = (const float*)d_in[1];
    const float* wl   = (const float*)d_in[2];
    const float* gl   = (const float*)d_in[3];
    const float* bl   = (const float*)d_in[4];
    const float* ml   = (const float*)d_in[5];
    const float* vl   = (const float*)d_in[6];
    const float* wg   = (const float*)d_in[7];
    const float* gg   = (const float*)d_in[8];
    const float* bg   = (const float*)d_in[9];
    const float* mg   = (const float*)d_in[10];
    const float* vg   = (const float*)d_in[11];

    float* out    = (float*)d_out;                                   // enhanced
    int*   idxout = (int*)((float*)d_out + (size_t)NB * CDIM * NPTS); // idx tail

    float4* pts = (float4*)d_ws;                                     // 512 KB
    float*  sc  = (float*)((char*)d_ws + (size_t)NB * NPTS * sizeof(float4));

    prep_pts_kernel<<<(NB * NPTS + 255) / 256, 256, 0, stream>>>(xyz, pts);
    prep_scales_kernel<<<2, 256, 0, stream>>>(gl, bl, ml, vl, gg, bg, mg, vg, sc);
    conv_fused_kernel<<<NB * 4 * (NPTS / 16), 256, 0, stream>>>(feat, wl, wg, sc, out);
    knn_wmma_kernel<<<NB * (NPTS / 16) / 8, 256, 0, stream>>>(pts, idxout);
}